// ClassAwareRegularization_75222057222988
// MI455X (gfx1250) — compile-verified
//
#include <hip/hip_runtime.h>

// Problem constants (match reference)
#define B_      8
#define D_      512
#define HW_     16384     // 128*128
#define C_      21
#define CP_     32        // padded class count (2 WMMA M-tiles)
#define TN_     256       // pixels per block tile
#define DT_     128       // d-rows staged per LDS stage (8 waves * 16)
#define STR_    260       // LDS row stride in floats (16B aligned, bank-spread)
#define NSTAGE_ (D_ / DT_)
#define IGNORE_ 255
#define EPS_    1e-6f

typedef __attribute__((ext_vector_type(2))) float v2f;
typedef __attribute__((ext_vector_type(8))) float v8f;

// D(16x16,f32) = A(16x4,f32) x B(4x16,f32) + C  -- CDNA5 V_WMMA_F32_16X16X4_F32
__device__ __forceinline__ v8f wmma_f32_4(v2f a, v2f b, v8f c) {
  return __builtin_amdgcn_wmma_f32_16x16x4_f32(false, a, false, b, (short)0, c,
                                               false, false);
}

// CDNA5 async DMA: global -> LDS, 16B per lane, tracked by ASYNCcnt.
// vdst = per-lane LDS byte address, vaddr = per-lane 64-bit global address.
__device__ __forceinline__ void async_g2l_b128(uint32_t lds_addr,
                                               const void* gaddr) {
  asm volatile("global_load_async_to_lds_b128 %0, %1, off"
               :: "v"(lds_addr), "v"((uint64_t)(uintptr_t)gaddr)
               : "memory");
}
__device__ __forceinline__ void wait_async_le(int prev) {
  if (prev) asm volatile("s_wait_asynccnt 0x20" ::: "memory");  // <= 32 left
  else      asm volatile("s_wait_asynccnt 0x0"  ::: "memory");  // all done
}

// -------- Pass 1: one streaming pass over features ------------------------
// Per block: batch b, 256-pixel tile, all D=512 channels.
// Double-buffered async staging overlapped with the WMMA segment-sum GEMM.
// Produces (atomic accumulate): S[b][c][d], counts[b][c], ssq = sum valid*|x|^2
extern "C" __global__ void __launch_bounds__(256)
car_pass1(const float* __restrict__ feat, const int* __restrict__ lab,
          float* __restrict__ S, float* __restrict__ counts,
          float* __restrict__ ssq)
{
  __shared__ __align__(16) float ldsF[2][DT_ * STR_];  // 2 x 130.0 KB
  __shared__ __align__(16) int   labs[TN_];
  __shared__ int hist[CP_];

  const int t    = threadIdx.x;
  const int b    = blockIdx.y;
  const int n0   = blockIdx.x * TN_;
  const int lane = t & 31;
  const int wave = t >> 5;
  const int m    = lane & 15;   // WMMA row/col within 16-tile
  const int kh   = lane >> 4;   // K-half select (lanes 16-31 hold K=2,3)

  const float* gbase = feat + (size_t)b * D_ * HW_ + n0;

  // issue stage 0 DMA immediately
  {
    #pragma unroll 4
    for (int j = 0; j < (DT_ * TN_ / 4) / 256; ++j) {
      const int idx = t + j * 256;
      const int row = idx >> 6;                 // 64 float4 per d-row
      const int c4  = idx & 63;
      async_g2l_b128((uint32_t)(uintptr_t)&ldsF[0][row * STR_ + c4 * 4],
                     gbase + (size_t)row * HW_ + c4 * 4);
    }
  }

  if (t < CP_) hist[t] = 0;
  labs[t] = lab[b * HW_ + n0 + t];
  __syncthreads();
  { int lv = labs[t]; if (lv != IGNORE_) atomicAdd(&hist[lv], 1); }

  float ss = 0.f;

  for (int s = 0; s < NSTAGE_; ++s) {
    // prefetch next stage into the other buffer (overlaps with compute)
    if (s + 1 < NSTAGE_) {
      const int nb = (s + 1) & 1;
      #pragma unroll 4
      for (int j = 0; j < (DT_ * TN_ / 4) / 256; ++j) {
        const int idx = t + j * 256;
        const int row = idx >> 6;
        const int c4  = idx & 63;
        async_g2l_b128((uint32_t)(uintptr_t)&ldsF[nb][row * STR_ + c4 * 4],
                       gbase + (size_t)((s + 1) * DT_ + row) * HW_ + c4 * 4);
      }
    }
    // oldest 32 async ops (= stage s) retired; barrier publishes all waves'
    wait_async_le(s + 1 < NSTAGE_);
    __syncthreads();

    // ---- per-wave 16(d) x 32(c) GEMM over K = 256 pixels, steps of 4 ----
    v8f acc0 = {};   // classes 0..15
    v8f acc1 = {};   // classes 16..31 (21..31 structurally zero)
    const float* bb = &ldsF[s & 1][(wave * 16 + m) * STR_];
    for (int ks = 0; ks < TN_ / 4; ++ks) {
      const int nl = ks * 4 + kh * 2;
      const int2 lp = *(const int2*)(labs + nl);   // labels for K pair
      v2f a0, a1, bf;
      a0.x = (lp.x == m)      ? 1.f : 0.f;   // one-hot (IGNORE never matches)
      a0.y = (lp.y == m)      ? 1.f : 0.f;
      a1.x = (lp.x == m + 16) ? 1.f : 0.f;
      a1.y = (lp.y == m + 16) ? 1.f : 0.f;
      const float2 bv = *(const float2*)(bb + nl); // B[k][d], d = lane&15
      bf.x = bv.x; bf.y = bv.y;
      // masked sum-of-squares: each (d,pixel) element visits exactly one lane
      ss += (lp.x != IGNORE_) ? bf.x * bf.x : 0.f;
      ss += (lp.y != IGNORE_) ? bf.y * bf.y : 0.f;
      acc0 = wmma_f32_4(a0, bf, acc0);
      acc1 = wmma_f32_4(a1, bf, acc1);
    }
    // ---- flush partial S tile: lane holds (c = r+8*kh [+16], d = m) ----
    const int dglob = s * DT_ + wave * 16 + m;
    #pragma unroll
    for (int r = 0; r < 8; ++r) {
      const int cc = r + 8 * kh;
      atomicAdd(&S[((size_t)b * CP_ + cc) * D_ + dglob], acc0[r]);
      if (cc + 16 < C_)
        atomicAdd(&S[((size_t)b * CP_ + cc + 16) * D_ + dglob], acc1[r]);
    }
    __syncthreads();   // buffer (s&1) free before stage s+2 DMA reuses it
  }

  // sum-of-squares: wave reduce, one atomic per wave
  for (int off = 16; off > 0; off >>= 1) ss += __shfl_xor(ss, off, 32);
  if (lane == 0) atomicAdd(ssq, ss);
  if (t < CP_ && hist[t] > 0) atomicAdd(&counts[b * CP_ + t], (float)hist[t]);
}

// -------- Pass 2: tiny epilogue (1 block) ---------------------------------
extern "C" __global__ void __launch_bounds__(256)
car_pass2(const float* __restrict__ S, const float* __restrict__ counts,
          const float* __restrict__ ssq, float* __restrict__ out)
{
  const int CMS = 516;                       // padded cm stride
  __shared__ float cm[C_ * 516];             // batch-mean centers
  __shared__ float inv[B_ * CP_];            // 1/(count+eps)
  __shared__ float Pl[B_ * C_];              // ||S_bc||^2
  __shared__ float Gs[32][32];               // Gram cm cm^T
  __shared__ float nrm[32];
  __shared__ float red[2];

  const int t = threadIdx.x;
  if (t < 2) red[t] = 0.f;
  if (t < B_ * CP_) inv[t] = 1.f / (counts[t] + EPS_);
  __syncthreads();

  // cm[c][d] = (1/B) sum_b S[b][c][d] / (n_bc + eps)
  for (int idx = t; idx < C_ * D_; idx += 256) {
    const int c = idx >> 9, d = idx & (D_ - 1);
    float a = 0.f;
    #pragma unroll
    for (int bb = 0; bb < B_; ++bb)
      a += S[((size_t)bb * CP_ + c) * D_ + d] * inv[bb * CP_ + c];
    cm[c * CMS + d] = a * (1.f / B_);
  }
  // Pl[b][c] = sum_d S^2
  if (t < B_ * C_) {
    const int bb = t / C_, c = t % C_;
    const float* sp = S + ((size_t)bb * CP_ + c) * D_;
    float p = 0.f;
    for (int d = 0; d < D_; ++d) { const float v = sp[d]; p += v * v; }
    Pl[t] = p;
  }
  __syncthreads();

  // Gram G = cm cm^T (32x32, K=512) with WMMA f32 16x16x4; wave 0 only.
  // A and B fragments coincide: lane holds row (lane&15), K pair 2*kh.
  if (t < 32) {
    const int m = t & 15, kh = t >> 4;
    const int c1 = m + 16;
    const float c1w = (c1 < C_) ? 1.f : 0.f;
    const int   c1i = (c1 < C_) ? c1 : 0;
    v8f g00 = {}, g01 = {}, g10 = {}, g11 = {};
    for (int k0 = 0; k0 < D_; k0 += 4) {
      const int k = k0 + kh * 2;
      v2f f0, f1;
      f0.x = cm[m * CMS + k];         f0.y = cm[m * CMS + k + 1];
      f1.x = cm[c1i * CMS + k] * c1w; f1.y = cm[c1i * CMS + k + 1] * c1w;
      g00 = wmma_f32_4(f0, f0, g00);
      g01 = wmma_f32_4(f0, f1, g01);
      g10 = wmma_f32_4(f1, f0, g10);
      g11 = wmma_f32_4(f1, f1, g11);
    }
    #pragma unroll
    for (int r = 0; r < 8; ++r) {
      const int mm = r + 8 * kh;
      Gs[mm][m]           = g00[r];
      Gs[mm][m + 16]      = g01[r];
      Gs[mm + 16][m]      = g10[r];
      Gs[mm + 16][m + 16] = g11[r];
    }
  }
  __syncthreads();
  if (t < C_) nrm[t] = sqrtf(fmaxf(Gs[t][t], 0.f));
  __syncthreads();

  // inter-class loss: sum_{c!=c'} relu(sim - 0.5)
  float accI = 0.f;
  for (int idx = t; idx < C_ * C_; idx += 256) {
    const int c = idx / C_, cp = idx % C_;
    if (c != cp) {
      const float s = Gs[c][cp] /
                      (fmaxf(nrm[c], 1e-12f) * fmaxf(nrm[cp], 1e-12f));
      const float r = s - 0.5f;
      accI += (r > 0.f) ? r : 0.f;
    }
  }
  atomicAdd(&red[1], accI);
  // intra: sum ||x-mu||^2 = Q + sum_bc P*(n*inv - 2)*inv
  if (t < B_ * C_) {
    const int bb = t / C_, c = t % C_;
    const float n  = counts[bb * CP_ + c];
    const float iv = inv[bb * CP_ + c];
    atomicAdd(&red[0], Pl[t] * (n * iv - 2.f) * iv);
  }
  __syncthreads();
  if (t == 0) {
    float vt = 0.f;
    for (int bb = 0; bb < B_; ++bb)
      for (int c = 0; c < C_; ++c) vt += counts[bb * CP_ + c];
    out[0] = (ssq[0] + red[0]) / (vt + EPS_);       // intra * INTRA_RATE(=1)
    out[1] = red[1] / ((float)C_ + EPS_);           // inter * INTER_RATE(=1)
  }
}

// -------- host launcher ---------------------------------------------------
extern "C" void kernel_launch(void* const* d_in, const int* in_sizes, int n_in,
                              void* d_out, int out_size, void* d_ws,
                              size_t ws_size, hipStream_t stream)
{
  const float* feat = (const float*)d_in[0];
  const int*   lab  = (const int*)d_in[1];
  float* out = (float*)d_out;

  float* S      = (float*)d_ws;                    // B*CP*D floats
  float* counts = S + (size_t)B_ * CP_ * D_;       // B*CP floats
  float* ssq    = counts + B_ * CP_;               // 1 float

  const size_t zbytes =
      ((size_t)B_ * CP_ * D_ + (size_t)B_ * CP_ + 1) * sizeof(float);
  hipMemsetAsync(d_ws, 0, zbytes, stream);

  dim3 grid1(HW_ / TN_, B_);
  car_pass1<<<grid1, 256, 0, stream>>>(feat, lab, S, counts, ssq);
  car_pass2<<<1, 256, 0, stream>>>(S, counts, ssq, out);
}